// HungarianLoss_87479893885054
// MI455X (gfx1250) — compile-verified
//
#include <hip/hip_runtime.h>
#include <math.h>

#define B_ 32
#define N_ 128
#define D_ 1024

typedef __attribute__((ext_vector_type(2))) float v2f;
typedef __attribute__((ext_vector_type(8))) float v8f;

// ---------------------------------------------------------------------------
// Kernel 1: per-row squared norms for pred (y==0) and target (y==1) nodes.
// grid = (B*N, 2), block = 256
// ---------------------------------------------------------------------------
__global__ void row_norms_kernel(const float* __restrict__ pred,
                                 const float* __restrict__ targ,
                                 float* __restrict__ normsA,
                                 float* __restrict__ normsB) {
    const float* src = (blockIdx.y == 0) ? pred : targ;
    float* dst       = (blockIdx.y == 0) ? normsA : normsB;
    int row = blockIdx.x;
    const float* p = src + (size_t)row * D_;
    float s = 0.f;
    for (int d = threadIdx.x; d < D_; d += blockDim.x) {
        float x = p[d];
        s += x * x;
    }
    __shared__ float red[256];
    red[threadIdx.x] = s;
    __syncthreads();
    for (int off = blockDim.x >> 1; off > 0; off >>= 1) {
        if (threadIdx.x < off) red[threadIdx.x] += red[threadIdx.x + off];
        __syncthreads();
    }
    if (threadIdx.x == 0) dst[row] = red[0];
}

// ---------------------------------------------------------------------------
// Kernel 2: cost[b][m][n] = sqrt(max(0, |a_m|^2 + |b_n|^2 - 2 a_m.b_n))
// GEMM part via V_WMMA_F32_16X16X4_F32 (fp32 in, fp32 acc).
// grid = (N/16, N/16, B), block = 32 (one wave, EXEC all ones)
// A layout (16x4 f32, 2 VGPRs/lane): lanes 0-15 -> M=lane, K={0,1};
//                                    lanes 16-31 -> M=lane-16, K={2,3}
// B layout mirrored with N instead of M.  C/D: VGPR r -> M = r + 8*(lane>=16),
// N = lane&15.
// ---------------------------------------------------------------------------
__global__ void cost_wmma_kernel(const float* __restrict__ pred,
                                 const float* __restrict__ targ,
                                 const float* __restrict__ normsA,
                                 const float* __restrict__ normsB,
                                 float* __restrict__ cost) {
    const int tile_n = blockIdx.x;
    const int tile_m = blockIdx.y;
    const int b      = blockIdx.z;
    const int lane   = threadIdx.x;
    const int half   = lane >> 4;   // 0: K 0..1 | 1: K 2..3 within the K4 window
    const int l16    = lane & 15;

    const int m = tile_m * 16 + l16;
    const int n = tile_n * 16 + l16;

    const v2f* pA = (const v2f*)(pred + ((size_t)(b * N_ + m)) * D_);
    const v2f* pB = (const v2f*)(targ + ((size_t)(b * N_ + n)) * D_);

    v8f acc = {};
#pragma unroll 8
    for (int k2 = 0; k2 < D_ / 2; k2 += 2) {   // k2 in float2 units; 4 K per step
        v2f a  = pA[k2 + half];
        v2f bb = pB[k2 + half];
        acc = __builtin_amdgcn_wmma_f32_16x16x4_f32(
            /*neg_a=*/false, a, /*neg_b=*/false, bb,
            /*c_mod=*/(short)0, acc, /*reuse_a=*/false, /*reuse_b=*/false);
    }

    const float nb = normsB[b * N_ + n];
    const int row_base = tile_m * 16 + half * 8;
#pragma unroll
    for (int r = 0; r < 8; ++r) {
        int row = row_base + r;
        float na  = normsA[b * N_ + row];
        float val = na + nb - 2.0f * acc[r];
        val = (val > 0.f) ? sqrtf(val) : 0.f;
        cost[((size_t)b * N_ + row) * N_ + (tile_n * 16 + l16)] = val;
    }
}

// ---------------------------------------------------------------------------
// Kernel 3: exact LAP (shortest augmenting path w/ potentials), one block
// per batch, 128 threads (thread t owns column j = t+1).  Cost matrix staged
// into LDS (64 KB dynamic shared; WGP has 320 KB).
// ---------------------------------------------------------------------------
__global__ void hungarian_kernel(const float* __restrict__ cost,
                                 int* __restrict__ col_out) {
    extern __shared__ float csh[];      // N_*N_ floats
    const int n = N_;
    const int b = blockIdx.x;
    const int tid = threadIdx.x;
    const int j = tid + 1;              // owned column (1-based)

    const float* cg = cost + (size_t)b * n * n;
    for (int k = tid; k < n * n; k += blockDim.x) csh[k] = cg[k];

    __shared__ float u[N_ + 1], v[N_ + 1], minv[N_ + 1];
    __shared__ int   p[N_ + 1], way[N_ + 1];
    __shared__ unsigned char used[N_ + 1];
    __shared__ float redv[N_];
    __shared__ int   redj[N_];
    __shared__ int   sj0, si0;

    u[j] = 0.f; v[j] = 0.f; p[j] = 0; way[j] = 0;
    if (tid == 0) { u[0] = 0.f; v[0] = 0.f; p[0] = 0; way[0] = 0; }
    __syncthreads();

    for (int i = 1; i <= n; ++i) {
        if (tid == 0) { p[0] = i; sj0 = 0; }
        minv[j] = 1e30f; used[j] = 0;
        if (tid == 0) { minv[0] = 1e30f; used[0] = 0; }
        __syncthreads();

        while (true) {
            if (tid == 0) { used[sj0] = 1; si0 = p[sj0]; }
            __syncthreads();
            const int i0 = si0;
            const float ui0 = u[i0];

            float bv = 1e30f;
            if (!used[j]) {
                float cand = csh[(i0 - 1) * n + (j - 1)] - ui0 - v[j];
                if (cand < minv[j]) { minv[j] = cand; way[j] = sj0; }
                bv = minv[j];
            }
            redv[tid] = bv; redj[tid] = j;
            __syncthreads();
            // tree argmin; strict '<' keeps the lower slot on ties (not always lowest j)
            for (int s = n >> 1; s > 0; s >>= 1) {
                if (tid < s && redv[tid + s] < redv[tid]) {
                    redv[tid] = redv[tid + s];
                    redj[tid] = redj[tid + s];
                }
                __syncthreads();
            }
            const int   j1    = redj[0];
            const float delta = redv[0];
            __syncthreads();

            if (used[j]) { u[p[j]] += delta; v[j] -= delta; }
            else         { minv[j] -= delta; }
            if (tid == 0) { u[p[0]] += delta; v[0] -= delta; sj0 = j1; }
            __syncthreads();

            if (p[sj0] == 0) break;
        }

        if (tid == 0) {            // augment along stored path
            int j0 = sj0;
            while (j0) { int jp = way[j0]; p[j0] = p[jp]; j0 = jp; }
        }
        __syncthreads();
    }
    // col_for_row[p[j]-1] = j-1
    col_out[b * n + (p[j] - 1)] = j - 1;
}

// ---------------------------------------------------------------------------
// Output zeroing (d_out is poisoned by the harness) and loss reductions.
// ---------------------------------------------------------------------------
__global__ void zero_out_kernel(float* __restrict__ out) {
    if (threadIdx.x < 2) out[threadIdx.x] = 0.f;
}

__global__ void node_loss_kernel(const float* __restrict__ pred,
                                 const float* __restrict__ targ,
                                 const int* __restrict__ col,
                                 float* __restrict__ out) {
    const size_t total = (size_t)B_ * N_ * D_;
    float s = 0.f;
    for (size_t idx = (size_t)blockIdx.x * blockDim.x + threadIdx.x; idx < total;
         idx += (size_t)gridDim.x * blockDim.x) {
        int d  = (int)(idx % D_);
        int bi = (int)(idx / D_);
        int i  = bi % N_;
        int b  = bi / N_;
        int tc = col[b * N_ + i];
        float diff = pred[idx] - targ[((size_t)(b * N_ + tc)) * D_ + d];
        s += diff * diff;
    }
    __shared__ float red[256];
    red[threadIdx.x] = s;
    __syncthreads();
    for (int off = blockDim.x >> 1; off > 0; off >>= 1) {
        if (threadIdx.x < off) red[threadIdx.x] += red[threadIdx.x + off];
        __syncthreads();
    }
    if (threadIdx.x == 0) atomicAdd(out, red[0] * (1.0f / (N_ * D_)));
}

__global__ void edge_loss_kernel(const float* __restrict__ pred_adj,
                                 const float* __restrict__ targ_adj,
                                 const int* __restrict__ col,
                                 float* __restrict__ out) {
    const size_t total = (size_t)B_ * N_ * N_;
    float s = 0.f;
    for (size_t idx = (size_t)blockIdx.x * blockDim.x + threadIdx.x; idx < total;
         idx += (size_t)gridDim.x * blockDim.x) {
        int jj = (int)(idx % N_);
        int t  = (int)(idx / N_);
        int ii = t % N_;
        int b  = t / N_;
        int ci = col[b * N_ + ii];
        int cj = col[b * N_ + jj];
        float diff = pred_adj[idx] - targ_adj[((size_t)b * N_ + ci) * N_ + cj];
        s += diff * diff;
    }
    __shared__ float red[256];
    red[threadIdx.x] = s;
    __syncthreads();
    for (int off = blockDim.x >> 1; off > 0; off >>= 1) {
        if (threadIdx.x < off) red[threadIdx.x] += red[threadIdx.x + off];
        __syncthreads();
    }
    if (threadIdx.x == 0) atomicAdd(out + 1, red[0] * (1.0f / (N_ * N_)));
}

// ---------------------------------------------------------------------------
extern "C" void kernel_launch(void* const* d_in, const int* in_sizes, int n_in,
                              void* d_out, int out_size, void* d_ws, size_t ws_size,
                              hipStream_t stream) {
    const float* pred_nodes   = (const float*)d_in[0];
    const float* target_nodes = (const float*)d_in[1];
    const float* pred_adj     = (const float*)d_in[2];
    const float* target_adj   = (const float*)d_in[3];
    float* out = (float*)d_out;

    // workspace layout: cost (B*N*N f32) | normsA (B*N) | normsB (B*N) | col (B*N i32)
    char*  ws     = (char*)d_ws;
    float* cost   = (float*)ws;
    float* normsA = (float*)(ws + (size_t)B_ * N_ * N_ * sizeof(float));
    float* normsB = normsA + B_ * N_;
    int*   col    = (int*)(normsB + B_ * N_);

    row_norms_kernel<<<dim3(B_ * N_, 2), 256, 0, stream>>>(
        pred_nodes, target_nodes, normsA, normsB);

    cost_wmma_kernel<<<dim3(N_ / 16, N_ / 16, B_), 32, 0, stream>>>(
        pred_nodes, target_nodes, normsA, normsB, cost);

    hungarian_kernel<<<B_, N_, N_ * N_ * sizeof(float), stream>>>(cost, col);

    zero_out_kernel<<<1, 32, 0, stream>>>(out);

    node_loss_kernel<<<1024, 256, 0, stream>>>(pred_nodes, target_nodes, col, out);
    edge_loss_kernel<<<256, 256, 0, stream>>>(pred_adj, target_adj, col, out);
}